// SMAL_89086211654343
// MI455X (gfx1250) — compile-verified
//
#include <hip/hip_runtime.h>
#include <hip/hip_bf16.h>
#include <math.h>

typedef __attribute__((ext_vector_type(2))) float v2f;
typedef __attribute__((ext_vector_type(8))) float v8f;

#define WMMA_F32(A, B, C) \
    __builtin_amdgcn_wmma_f32_16x16x4_f32(false, (A), false, (B), (short)0, (C), false, false)

// ---------------------------------------------------------------------------
// Kernel 1: v_shaped[b, x] = v_template[x] + sum_k beta[b,k] * shapedirs[k,x]
// GEMM M=B, N=V*3, K=NB(30). One wave32 per 16x32 tile (two 16x16 WMMA tiles).
// Loads are unconditional from clamped addresses; only K-tail uses selects.
// ---------------------------------------------------------------------------
__global__ void k_shape_blend(const float* __restrict__ beta,       // (B, K)
                              const float* __restrict__ shapedirs,  // (K, N)
                              const float* __restrict__ v_template, // (N)
                              float* __restrict__ v_shaped,         // (B, N)
                              int Bn, int N, int K)
{
    const int tn = blockIdx.x;          // 32-wide N tile
    const int tm = blockIdx.y;          // 16-wide M (batch) tile
    const int lane = threadIdx.x;
    const int l = lane & 15;
    const int h = lane >> 4;
    const int kb = 2 * h;

    const int row  = tm * 16 + l;
    const int rowc = row < Bn ? row : Bn - 1;
    const int col0 = tn * 32 + l;
    const int col1 = col0 + 16;
    const int c0c  = col0 < N ? col0 : N - 1;
    const int c1c  = col1 < N ? col1 : N - 1;

    const float* __restrict__ Ap = beta + (size_t)rowc * K;
    const float* __restrict__ B0 = shapedirs + c0c;
    const float* __restrict__ B1 = shapedirs + c1c;

    v8f acc0 = {}, acc1 = {};
    int k0 = 0;
    #pragma unroll 2
    for (; k0 + 4 <= K; k0 += 4) {
        const int ka = k0 + kb;
        v2f a, b0, b1;
        a.x  = Ap[ka];
        a.y  = Ap[ka + 1];
        b0.x = B0[(size_t)ka * N];
        b0.y = B0[(size_t)(ka + 1) * N];
        b1.x = B1[(size_t)ka * N];
        b1.y = B1[(size_t)(ka + 1) * N];
        acc0 = WMMA_F32(a, b0, acc0);
        acc1 = WMMA_F32(a, b1, acc1);
    }
    if (k0 < K) {                       // K tail (K not multiple of 4)
        const int  ka  = k0 + kb;
        const bool v0  = (ka     < K);
        const bool v1  = (ka + 1 < K);
        const int  ka0 = v0 ? ka     : K - 1;
        const int  ka1 = v1 ? ka + 1 : K - 1;
        v2f a, b0, b1;
        float t;
        t = Ap[ka0];               a.x  = v0 ? t : 0.0f;
        t = Ap[ka1];               a.y  = v1 ? t : 0.0f;
        t = B0[(size_t)ka0 * N];   b0.x = v0 ? t : 0.0f;
        t = B0[(size_t)ka1 * N];   b0.y = v1 ? t : 0.0f;
        t = B1[(size_t)ka0 * N];   b1.x = v0 ? t : 0.0f;
        t = B1[(size_t)ka1 * N];   b1.y = v1 ? t : 0.0f;
        acc0 = WMMA_F32(a, b0, acc0);
        acc1 = WMMA_F32(a, b1, acc1);
    }

    if (col0 < N) {
        const float vt = v_template[col0];
        #pragma unroll
        for (int r = 0; r < 8; ++r) {
            const int m = tm * 16 + r + 8 * h;
            if (m < Bn) v_shaped[(size_t)m * N + col0] = acc0[r] + vt;
        }
    }
    if (col1 < N) {
        const float vt = v_template[col1];
        #pragma unroll
        for (int r = 0; r < 8; ++r) {
            const int m = tm * 16 + r + 8 * h;
            if (m < Bn) v_shaped[(size_t)m * N + col1] = acc1[r] + vt;
        }
    }
}

// ---------------------------------------------------------------------------
// Kernel 2: Jloc[b,j,c] = sum_v v_shaped[b, v*3+c] * Jreg[v,j]
// ---------------------------------------------------------------------------
__global__ void k_jloc(const float* __restrict__ v_shaped, // (B, V*3)
                       const float* __restrict__ Jreg,     // (V, J)
                       float* __restrict__ Jloc,           // (B, J, 3)
                       int Vn, int Jn)
{
    const int b = blockIdx.x / Jn;
    const int j = blockIdx.x % Jn;
    const float* vs = v_shaped + (size_t)b * Vn * 3;

    float s0 = 0.f, s1 = 0.f, s2 = 0.f;
    for (int v = threadIdx.x; v < Vn; v += 32) {
        const float w = Jreg[(size_t)v * Jn + j];
        s0 += w * vs[v * 3 + 0];
        s1 += w * vs[v * 3 + 1];
        s2 += w * vs[v * 3 + 2];
    }
    #pragma unroll
    for (int off = 16; off > 0; off >>= 1) {
        s0 += __shfl_xor(s0, off, 32);
        s1 += __shfl_xor(s1, off, 32);
        s2 += __shfl_xor(s2, off, 32);
    }
    if (threadIdx.x == 0) {
        float* o = Jloc + ((size_t)b * Jn + j) * 3;
        o[0] = s0; o[1] = s1; o[2] = s2;
    }
}

// ---------------------------------------------------------------------------
// Kernel 3: s[b,m] = exp( sum_l betas_limbs[b,l] * mask[l,m] )
// ---------------------------------------------------------------------------
__global__ void k_scale(const float* __restrict__ bl,   // (B, NL)
                        const float* __restrict__ mask, // (NL, M)
                        float* __restrict__ sbuf,       // (B, M)
                        int Bn, int NLn, int M)
{
    const int idx = blockIdx.x * blockDim.x + threadIdx.x;
    if (idx >= Bn * M) return;
    const int b = idx / M;
    const int m = idx % M;
    float acc = 0.f;
    for (int l = 0; l < NLn; ++l)
        acc += bl[b * NLn + l] * mask[l * M + m];
    sbuf[idx] = expf(acc);
}

// ---------------------------------------------------------------------------
// Kernel 4: kinematic chain. One thread per batch (sequential over J).
// ---------------------------------------------------------------------------
__global__ void k_chain(const float* __restrict__ pose,    // (B, J*9)
                        const float* __restrict__ Jloc,    // (B, J, 3)
                        const float* __restrict__ sbuf,    // (B, J*3)
                        const int*   __restrict__ parents, // (J)
                        float* __restrict__ Aout,          // (B, J, 16)
                        int Bn, int Jn)
{
    const int b = blockIdx.x * blockDim.x + threadIdx.x;
    if (b >= Bn) return;

    const float* P  = pose + (size_t)b * Jn * 9;
    const float* JL = Jloc + (size_t)b * Jn * 3;
    const float* S  = sbuf + (size_t)b * Jn * 3;

    float res[35][16];

    {
        float* R = res[0];
        for (int r = 0; r < 3; ++r) {
            for (int c = 0; c < 3; ++c) R[r * 4 + c] = P[r * 3 + c];
            R[r * 4 + 3] = JL[r];
        }
        R[12] = 0.f; R[13] = 0.f; R[14] = 0.f; R[15] = 1.f;
    }

    for (int i = 1; i < Jn; ++i) {
        const int p = parents[i];
        float Ai[16];
        for (int r = 0; r < 3; ++r) {
            for (int c = 0; c < 3; ++c)
                Ai[r * 4 + c] = P[i * 9 + r * 3 + c] * S[i * 3 + c] / S[p * 3 + r];
            Ai[r * 4 + 3] = JL[i * 3 + r] - JL[p * 3 + r];
        }
        Ai[12] = 0.f; Ai[13] = 0.f; Ai[14] = 0.f; Ai[15] = 1.f;
        for (int r = 0; r < 4; ++r)
            for (int c = 0; c < 4; ++c) {
                float acc = 0.f;
                for (int k = 0; k < 4; ++k) acc += res[p][r * 4 + k] * Ai[k * 4 + c];
                res[i][r * 4 + c] = acc;
            }
    }

    for (int j = 0; j < Jn; ++j) {
        const float jx = JL[j * 3 + 0], jy = JL[j * 3 + 1], jz = JL[j * 3 + 2];
        for (int r = 0; r < 4; ++r) {
            const float ib = res[j][r * 4 + 0] * jx + res[j][r * 4 + 1] * jy +
                             res[j][r * 4 + 2] * jz;
            res[j][r * 4 + 3] -= ib;
        }
        float* o = Aout + ((size_t)b * Jn + j) * 16;
        for (int q = 0; q < 16; ++q) o[q] = res[j][q];
    }
}

// ---------------------------------------------------------------------------
// Kernel 5: v_posed = v_shaped + pose_feature @ posedirs  (in-place on ws)
// pose_feature computed on the fly from pose.  K=(J-1)*9=306.
// ---------------------------------------------------------------------------
__global__ void k_pose_blend(const float* __restrict__ pose,     // (B, J*9)
                             const float* __restrict__ posedirs, // (K, N)
                             float* __restrict__ v_inout,        // (B, N)
                             int Bn, int N, int Jn)
{
    const int K = (Jn - 1) * 9;
    const int tn = blockIdx.x;
    const int tm = blockIdx.y;
    const int lane = threadIdx.x;
    const int l = lane & 15;
    const int h = lane >> 4;
    const int kb = 2 * h;

    const int row  = tm * 16 + l;
    const int rowc = row < Bn ? row : Bn - 1;
    const int col0 = tn * 32 + l;
    const int col1 = col0 + 16;
    const int c0c  = col0 < N ? col0 : N - 1;
    const int c1c  = col1 < N ? col1 : N - 1;

    const float* __restrict__ Pp = pose + (size_t)rowc * Jn * 9 + 9; // pose[b,1:,..]
    const float* __restrict__ B0 = posedirs + c0c;
    const float* __restrict__ B1 = posedirs + c1c;

    v8f acc0 = {}, acc1 = {};
    int k0 = 0;
    #pragma unroll 2
    for (; k0 + 4 <= K; k0 += 4) {
        const int ka = k0 + kb;
        const int m0 = ka % 9;
        const int m1 = (ka + 1) % 9;
        v2f a, b0, b1;
        a.x  = Pp[ka]     - (((m0 & 3) == 0) ? 1.0f : 0.0f);
        a.y  = Pp[ka + 1] - (((m1 & 3) == 0) ? 1.0f : 0.0f);
        b0.x = B0[(size_t)ka * N];
        b0.y = B0[(size_t)(ka + 1) * N];
        b1.x = B1[(size_t)ka * N];
        b1.y = B1[(size_t)(ka + 1) * N];
        acc0 = WMMA_F32(a, b0, acc0);
        acc1 = WMMA_F32(a, b1, acc1);
    }
    if (k0 < K) {                       // K tail
        const int  ka  = k0 + kb;
        const bool v0  = (ka     < K);
        const bool v1  = (ka + 1 < K);
        const int  ka0 = v0 ? ka     : K - 1;
        const int  ka1 = v1 ? ka + 1 : K - 1;
        const int  m0  = ka0 % 9;
        const int  m1  = ka1 % 9;
        v2f a, b0, b1;
        float t;
        t = Pp[ka0] - (((m0 & 3) == 0) ? 1.0f : 0.0f);  a.x  = v0 ? t : 0.0f;
        t = Pp[ka1] - (((m1 & 3) == 0) ? 1.0f : 0.0f);  a.y  = v1 ? t : 0.0f;
        t = B0[(size_t)ka0 * N];                        b0.x = v0 ? t : 0.0f;
        t = B0[(size_t)ka1 * N];                        b0.y = v1 ? t : 0.0f;
        t = B1[(size_t)ka0 * N];                        b1.x = v0 ? t : 0.0f;
        t = B1[(size_t)ka1 * N];                        b1.y = v1 ? t : 0.0f;
        acc0 = WMMA_F32(a, b0, acc0);
        acc1 = WMMA_F32(a, b1, acc1);
    }

    if (col0 < N) {
        #pragma unroll
        for (int r = 0; r < 8; ++r) {
            const int m = tm * 16 + r + 8 * h;
            if (m < Bn) {
                const size_t idx = (size_t)m * N + col0;
                v_inout[idx] = v_inout[idx] + acc0[r];
            }
        }
    }
    if (col1 < N) {
        #pragma unroll
        for (int r = 0; r < 8; ++r) {
            const int m = tm * 16 + r + 8 * h;
            if (m < Bn) {
                const size_t idx = (size_t)m * N + col1;
                v_inout[idx] = v_inout[idx] + acc1[r];
            }
        }
    }
}

// ---------------------------------------------------------------------------
// Kernel 6: skinning.  Per (b, 16-vertex tile): T(16x16) = W(16x35) @ A_b(35x16)
// via WMMA, spill to LDS, lanes 0..15 apply T * homo + trans.
// ---------------------------------------------------------------------------
__global__ void k_skin(const float* __restrict__ weights, // (V, J)
                       const float* __restrict__ Abuf,    // (B, J, 16)
                       const float* __restrict__ v_posed, // (B, V*3)
                       const float* __restrict__ trans,   // (B, 3)
                       float* __restrict__ verts,         // (B, V, 3)
                       int Bn, int Vn, int Jn)
{
    __shared__ float Tt[16][16];
    const int tv = blockIdx.x;
    const int b  = blockIdx.y;
    const int lane = threadIdx.x;
    const int l = lane & 15;
    const int h = lane >> 4;
    const int kb = 2 * h;

    const int vrow  = tv * 16 + l;
    const int vrowc = vrow < Vn ? vrow : Vn - 1;
    const float* __restrict__ Wp = weights + (size_t)vrowc * Jn;
    const float* __restrict__ Ab = Abuf + (size_t)b * Jn * 16 + l;

    v8f acc = {};
    int k0 = 0;
    #pragma unroll 2
    for (; k0 + 4 <= Jn; k0 += 4) {
        const int ka = k0 + kb;
        v2f a, bb;
        a.x  = Wp[ka];
        a.y  = Wp[ka + 1];
        bb.x = Ab[(ka)     * 16];
        bb.y = Ab[(ka + 1) * 16];
        acc = WMMA_F32(a, bb, acc);
    }
    if (k0 < Jn) {                      // K tail (Jn=35)
        const int  ka  = k0 + kb;
        const bool v0  = (ka     < Jn);
        const bool v1  = (ka + 1 < Jn);
        const int  ka0 = v0 ? ka     : Jn - 1;
        const int  ka1 = v1 ? ka + 1 : Jn - 1;
        v2f a, bb;
        float t;
        t = Wp[ka0];       a.x  = v0 ? t : 0.0f;
        t = Wp[ka1];       a.y  = v1 ? t : 0.0f;
        t = Ab[ka0 * 16];  bb.x = v0 ? t : 0.0f;
        t = Ab[ka1 * 16];  bb.y = v1 ? t : 0.0f;
        acc = WMMA_F32(a, bb, acc);
    }

    #pragma unroll
    for (int r = 0; r < 8; ++r) Tt[r + 8 * h][l] = acc[r];
    __syncthreads();

    if (h == 0) {
        const int v = tv * 16 + l;
        if (v < Vn) {
            const float* vp = v_posed + (size_t)b * Vn * 3 + (size_t)v * 3;
            const float hx = vp[0], hy = vp[1], hz = vp[2];
            float* o = verts + ((size_t)b * Vn + v) * 3;
            #pragma unroll
            for (int c = 0; c < 3; ++c) {
                o[c] = Tt[l][c * 4 + 0] * hx + Tt[l][c * 4 + 1] * hy +
                       Tt[l][c * 4 + 2] * hz + Tt[l][c * 4 + 3] + trans[b * 3 + c];
            }
        }
    }
}

// ---------------------------------------------------------------------------
extern "C" void kernel_launch(void* const* d_in, const int* in_sizes, int n_in,
                              void* d_out, int out_size, void* d_ws, size_t ws_size,
                              hipStream_t stream)
{
    const float* beta       = (const float*)d_in[0];
    const float* bl         = (const float*)d_in[1];
    const float* pose       = (const float*)d_in[2];
    const float* trans      = (const float*)d_in[3];
    const float* v_template = (const float*)d_in[4];
    const float* shapedirs  = (const float*)d_in[5];
    const float* posedirs   = (const float*)d_in[6];
    const float* Jreg       = (const float*)d_in[7];
    const float* weights    = (const float*)d_in[8];
    const float* mask       = (const float*)d_in[9];
    const int*   parents    = (const int*)d_in[10];
    float* verts = (float*)d_out;

    const int Jn  = in_sizes[10];          // 35
    const int Bn  = in_sizes[3] / 3;       // 1024
    const int Vn  = in_sizes[4] / 3;       // 3889
    const int NBn = in_sizes[0] / Bn;      // 30
    const int NLn = in_sizes[1] / Bn;      // 7
    const int N   = Vn * 3;                // 11667

    float* ws     = (float*)d_ws;
    float* vsh    = ws;                                  // (B, N)
    float* Jloc   = vsh  + (size_t)Bn * N;               // (B, J, 3)
    float* sbuf   = Jloc + (size_t)Bn * Jn * 3;          // (B, J*3)
    float* Abuf   = sbuf + (size_t)Bn * Jn * 3;          // (B, J, 16)

    const int tilesM  = (Bn + 15) / 16;
    const int tilesN2 = (N + 31) / 32;      // 32-wide tiles (NT=2)
    const int tilesV  = (Vn + 15) / 16;

    k_shape_blend<<<dim3(tilesN2, tilesM), 32, 0, stream>>>(
        beta, shapedirs, v_template, vsh, Bn, N, NBn);

    k_jloc<<<Bn * Jn, 32, 0, stream>>>(vsh, Jreg, Jloc, Vn, Jn);

    {
        const int total = Bn * Jn * 3;
        k_scale<<<(total + 255) / 256, 256, 0, stream>>>(bl, mask, sbuf, Bn, NLn, Jn * 3);
    }

    k_chain<<<(Bn + 255) / 256, 256, 0, stream>>>(pose, Jloc, sbuf, parents, Abuf, Bn, Jn);

    k_pose_blend<<<dim3(tilesN2, tilesM), 32, 0, stream>>>(pose, posedirs, vsh, Bn, N, Jn);

    k_skin<<<dim3(tilesV, Bn), 32, 0, stream>>>(weights, Abuf, vsh, trans, verts, Bn, Vn, Jn);
}